// GNN_35828617183700
// MI455X (gfx1250) — compile-verified
//
#include <hip/hip_runtime.h>
#include <hip/hip_bf16.h>

#define H_DIM 32
#define DIN_DIM 384
#define N_LAYERS 3
#define BN_EPS 1e-5f

typedef __attribute__((ext_vector_type(2))) float v2f;
typedef __attribute__((ext_vector_type(8))) float v8f;

// ---------------- small utility kernels ----------------

__global__ void zero_f32(float* __restrict__ p, int n) {
  int i = blockIdx.x * blockDim.x + threadIdx.x;
  if (i < n) p[i] = 0.0f;
}

__global__ void deg_init(int* __restrict__ deg, int n) {
  int i = blockIdx.x * blockDim.x + threadIdx.x;
  if (i < n) deg[i] = 1;                  // self-loop contributes 1
}

__global__ void deg_count(const long long* __restrict__ ei, int E, int* __restrict__ deg) {
  int i = blockIdx.x * blockDim.x + threadIdx.x;
  int stride = gridDim.x * blockDim.x;
  for (int e = i; e < E; e += stride)
    atomicAdd(&deg[(int)ei[e]], 1);       // dst = edge_index[0]
}

__global__ void deg_fin(const int* __restrict__ deg, float* __restrict__ dinv,
                        float* __restrict__ invc, int n) {
  int i = blockIdx.x * blockDim.x + threadIdx.x;
  if (i < n) {
    float d = (float)deg[i];
    dinv[i] = rsqrtf(d);
    invc[i] = 1.0f / d;
  }
}

// ---------------- WMMA GEMM: z = A[nrows,K](lda) @ B[K,32] ----------------
// Epilogue also writes acc = dinv[row]^2 * z  (self-loop init for scatter).
// One wave computes a 16x32 output tile: two f32 WMMA accumulators sharing A.
__global__ void gemm_wmma(const float* __restrict__ A, int lda, int K,
                          const float* __restrict__ B,
                          const float* __restrict__ dinv,
                          float* __restrict__ z, float* __restrict__ acc, int nrows) {
  int wid  = (blockIdx.x * blockDim.x + threadIdx.x) >> 5;  // wave-uniform
  int lane = threadIdx.x & 31;
  int m0 = wid * 16;
  if (m0 >= nrows) return;                                   // uniform exit, EXEC all-1 below
  int half = lane >> 4;          // 0: lanes 0-15, 1: lanes 16-31
  int l16  = lane & 15;

  int mrow = m0 + l16;
  if (mrow > nrows - 1) mrow = nrows - 1;                    // clamp loads (stores guarded)
  // A 16x4 f32 fragment: lanes 0-15 hold K=0,1; lanes 16-31 hold K=2,3 (M = l16)
  const float* arow  = A + (size_t)mrow * lda + 2 * half;
  // B 4x16 f32 fragment: lanes 0-15 hold rows K=0,1; lanes 16-31 rows K=2,3 (N = l16)
  const float* bbase = B + 2 * half * H_DIM + l16;

  v8f c0 = {0.f,0.f,0.f,0.f,0.f,0.f,0.f,0.f};
  v8f c1 = {0.f,0.f,0.f,0.f,0.f,0.f,0.f,0.f};
  for (int k = 0; k < K; k += 4) {
    v2f a;  a.x  = arow[k];      a.y  = arow[k + 1];
    const float* bp = bbase + (size_t)k * H_DIM;
    v2f b0; b0.x = bp[0];        b0.y = bp[H_DIM];           // N tile 0..15
    v2f b1; b1.x = bp[16];       b1.y = bp[H_DIM + 16];      // N tile 16..31
    c0 = __builtin_amdgcn_wmma_f32_16x16x4_f32(false, a, false, b0, (short)0, c0, false, false);
    c1 = __builtin_amdgcn_wmma_f32_16x16x4_f32(false, a, false, b1, (short)0, c1, false, false);
  }

  // D layout: VGPR j -> row m0+j (lanes 0-15) / m0+j+8 (lanes 16-31), col = l16
#pragma unroll
  for (int j = 0; j < 8; ++j) {
    int row = m0 + j + 8 * half;
    if (row < nrows) {
      float dv = dinv[row];
      float s  = dv * dv;
      size_t o = (size_t)row * H_DIM + l16;
      float v0 = c0[j], v1 = c1[j];
      z[o]        = v0;       z[o + 16]   = v1;
      acc[o]      = s * v0;   acc[o + 16] = s * v1;
    }
  }
}

// ---------------- edge gather/scatter: wave per edge, lane per feature ----------------
__global__ void edge_scatter(const long long* __restrict__ ei, int E,
                             const float* __restrict__ z,
                             const float* __restrict__ dinv,
                             float* __restrict__ acc) {
  int wid  = (blockIdx.x * blockDim.x + threadIdx.x) >> 5;
  int nw   = (gridDim.x * blockDim.x) >> 5;
  int lane = threadIdx.x & 31;
  for (int e = wid; e < E; e += nw) {
    int dst = (int)ei[e];
    int src = (int)ei[(size_t)E + e];
    float w = dinv[dst] * dinv[src];                 // symmetric GCN norm
    float v = z[(size_t)src * H_DIM + lane];         // coalesced 128B row
    atomicAdd(&acc[(size_t)dst * H_DIM + lane], w * v);
  }
}

// ---------------- mean-divide + bias, accumulate BN sum/sumsq ----------------
__global__ void finalize_stats(const float* __restrict__ acc, const float* __restrict__ invc,
                               const float* __restrict__ bias, float* __restrict__ z2,
                               float* __restrict__ stats, int n) {
  __shared__ float s1[256];
  __shared__ float s2[256];
  int t = threadIdx.x;
  int f = t & 31;                       // stride is multiple of 32 -> fixed feature per thread
  float sum = 0.f, sq = 0.f;
  int total = n * H_DIM;
  int stride = gridDim.x * blockDim.x;
  for (int i = blockIdx.x * blockDim.x + t; i < total; i += stride) {
    int row = i >> 5;
    float v = acc[i] * invc[row] + bias[f];
    z2[i] = v;
    sum += v; sq += v * v;
  }
  s1[t] = sum; s2[t] = sq;
  __syncthreads();
  for (int st = 128; st >= 32; st >>= 1) {
    if (t < st) { s1[t] += s1[t + st]; s2[t] += s2[t + st]; }
    __syncthreads();
  }
  if (t < 32) {
    atomicAdd(&stats[f],      s1[t]);
    atomicAdd(&stats[32 + f], s2[t]);
  }
}

__global__ void bn_params(const float* __restrict__ stats, const float* __restrict__ gamma,
                          const float* __restrict__ beta, float* __restrict__ scale,
                          float* __restrict__ shift, float invn) {
  int f = threadIdx.x;                  // 32 threads
  float mu   = stats[f] * invn;
  float var  = stats[32 + f] * invn - mu * mu;
  float rstd = rsqrtf(var + BN_EPS);
  float g    = rstd * gamma[f];
  scale[f] = g;
  shift[f] = beta[f] - mu * g;
}

__global__ void bn_relu(const float* __restrict__ z2, const float* __restrict__ scale,
                        const float* __restrict__ shift, float* __restrict__ h, int total) {
  int stride = gridDim.x * blockDim.x;
  for (int i = blockIdx.x * blockDim.x + threadIdx.x; i < total; i += stride) {
    int f = i & 31;
    float v = z2[i] * scale[f] + shift[f];
    h[i] = fmaxf(v, 0.0f);
  }
}

// ---------------- global mean pool + linear head ----------------
__global__ void pool_kernel(const float* __restrict__ h, const long long* __restrict__ batch,
                            float* __restrict__ psum, float* __restrict__ cnt, int n) {
  int stride = gridDim.x * blockDim.x;
  int total = n * H_DIM;
  for (int i = blockIdx.x * blockDim.x + threadIdx.x; i < total; i += stride) {
    int row = i >> 5, f = i & 31;
    int g = (int)batch[row];
    atomicAdd(&psum[g * H_DIM + f], h[i]);
    if (f == 0) atomicAdd(&cnt[g], 1.0f);
  }
}

__global__ void head_kernel(const float* __restrict__ psum, const float* __restrict__ cnt,
                            const float* __restrict__ headW, const float* __restrict__ headb,
                            float* __restrict__ out, int G) {
  int idx = blockIdx.x * blockDim.x + threadIdx.x;
  if (idx < G * 2) {
    int g = idx >> 1, c = idx & 1;
    float ic = 1.0f / fmaxf(cnt[g], 1.0f);
    float s = 0.f;
#pragma unroll
    for (int f = 0; f < H_DIM; ++f) s += psum[g * H_DIM + f] * headW[f * 2 + c];
    out[idx] = s * ic + headb[c];
  }
}

// ---------------- host orchestration ----------------
extern "C" void kernel_launch(void* const* d_in, const int* in_sizes, int n_in,
                              void* d_out, int out_size, void* d_ws, size_t ws_size,
                              hipStream_t stream) {
  const float*     x     = (const float*)d_in[0];
  const long long* ei    = (const long long*)d_in[1];   // [2,E] int64
  const long long* batch = (const long long*)d_in[3];   // [N] int64 (edge_attr d_in[2] unused)
  const float* W0    = (const float*)d_in[4];
  const float* Wr    = (const float*)d_in[5];
  const float* bias  = (const float*)d_in[6];
  const float* gamma = (const float*)d_in[7];
  const float* beta  = (const float*)d_in[8];
  const float* headW = (const float*)d_in[9];
  const float* headb = (const float*)d_in[10];
  float* out = (float*)d_out;

  const int N = in_sizes[0] / DIN_DIM;
  const int E = in_sizes[1] / 2;
  const int G = out_size / 2;
  (void)n_in; (void)ws_size;

  // workspace carve-up (256B aligned slices)
  char* wp = (char*)d_ws;
  auto carve = [&](size_t bytes) -> char* {
    char* p = wp; wp += (bytes + 255) & ~(size_t)255; return p;
  };
  int*   deg   = (int*)  carve((size_t)N * 4);
  float* dinv  = (float*)carve((size_t)N * 4);
  float* invc  = (float*)carve((size_t)N * 4);
  float* z     = (float*)carve((size_t)N * H_DIM * 4);   // z, reused as z2
  float* acc   = (float*)carve((size_t)N * H_DIM * 4);
  float* hbuf  = (float*)carve((size_t)N * H_DIM * 4);
  float* stats = (float*)carve(64 * 4);
  float* scale = (float*)carve(32 * 4);
  float* shift = (float*)carve(32 * 4);
  float* psum  = (float*)carve((size_t)(G * H_DIM + G) * 4);
  float* cnt   = psum + (size_t)G * H_DIM;

  dim3 blk(256);
  int nblkN = (N + 255) / 256;

  // degrees (with self-loops), dinv, 1/deg
  deg_init<<<nblkN, blk, 0, stream>>>(deg, N);
  deg_count<<<2048, blk, 0, stream>>>(ei, E, deg);
  deg_fin<<<nblkN, blk, 0, stream>>>(deg, dinv, invc, N);

  const float* A = x; int lda = DIN_DIM, K = DIN_DIM;
  for (int l = 0; l < N_LAYERS; ++l) {
    const float* Wl = (l == 0) ? W0 : (Wr + (size_t)(l - 1) * H_DIM * H_DIM);
    int mtiles  = (N + 15) / 16;
    int gblocks = (mtiles + 7) / 8;     // 8 waves (16-row tiles) per block
    gemm_wmma<<<gblocks, blk, 0, stream>>>(A, lda, K, Wl, dinv, z, acc, N);
    edge_scatter<<<4096, blk, 0, stream>>>(ei, E, z, dinv, acc);
    zero_f32<<<1, 64, 0, stream>>>(stats, 64);
    finalize_stats<<<2048, blk, 0, stream>>>(acc, invc, bias + l * H_DIM, z, stats, N);
    bn_params<<<1, 32, 0, stream>>>(stats, gamma + l * H_DIM, beta + l * H_DIM,
                                    scale, shift, 1.0f / (float)N);
    bn_relu<<<2048, blk, 0, stream>>>(z, scale, shift, hbuf, N * H_DIM);
    A = hbuf; lda = H_DIM; K = H_DIM;
  }

  int pz = G * H_DIM + G;
  zero_f32<<<(pz + 255) / 256, blk, 0, stream>>>(psum, pz);
  pool_kernel<<<2048, blk, 0, stream>>>(hbuf, batch, psum, cnt, N);
  head_kernel<<<1, 256, 0, stream>>>(psum, cnt, headW, headb, out, G);
}